// IntShear4_26405458936217
// MI455X (gfx1250) — compile-verified
//
#include <hip/hip_runtime.h>

// CDNA5 / gfx1250, wave32. One wave does the whole job:
//   M = S_11 @ ... @ S_1 @ S_0,  S_p = I + k_int[p] * e_i e_j^T  (4x4)
// Each chain step is one V_WMMA_F32_16X16X4_F32 (A=16x4 = S_p, B=4x16 = M).
// All state lives in VGPRs/SGPRs: scalars broadcast with v_readlane, the
// D(C/D-layout) -> B(layout) conversion between steps uses ds_bpermute_b32
// (lane permute only -- no LDS memory, no barriers, EXEC stays all-1s).

typedef __attribute__((ext_vector_type(2))) float v2f;
typedef __attribute__((ext_vector_type(8))) float v8f;

__device__ __forceinline__ float bperm_f(int srcLane, float v) {
    // ds_bpermute_b32: gather, byte index (lane*4); wave32 uses index[6:2]
    return __int_as_float(
        __builtin_amdgcn_ds_bpermute(srcLane << 2, __float_as_int(v)));
}

__global__ __launch_bounds__(32) void intshear4_wmma_kernel(
    const float* __restrict__ k_raw,
    const int*   __restrict__ pairs_i,
    const int*   __restrict__ pairs_j,
    float*       __restrict__ out)
{
    const int lane = threadIdx.x;          // 0..31 (wave32)

    // Lane p (p < 12) holds step p's scalars in registers.
    float kv_all = 0.0f;
    int   ip_all = 0, jp_all = 0;
    if (lane < 12) {
        kv_all = rintf(3.0f * tanhf(k_raw[lane]));  // round(K_MAX*tanh), RNE
        ip_all = pairs_i[lane];
        jp_all = pairs_j[lane];
    }

    // Static per-lane coordinates for the WMMA f32 16x16x4 operand layouts.
    // A (16x4): lane -> row M (lane&15); lanes 0-15 hold K=0,1; 16-31 hold K=2,3.
    // B (4x16): lane -> col N (lane&15); v0/v1 hold K rows k0/k1.
    const int  mrow = lane & 15;
    const int  k0   = (lane >> 4) << 1;
    const int  k1   = k0 + 1;
    const int  ncol = lane & 15;
    const bool lo   = (lane < 16);

    // B = M = I embedded in the 4x16 tile (cols >= 4 zero).
    v2f b;
    b.x = (ncol == k0) ? 1.0f : 0.0f;
    b.y = (ncol == k1) ? 1.0f : 0.0f;

    v8f d = {};
#pragma unroll
    for (int p = 0; p < 12; ++p) {
        // Uniform broadcast of step-p scalars: v_readlane -> SGPRs.
        const float kv =
            __int_as_float(__builtin_amdgcn_readlane(__float_as_int(kv_all), p));
        const int ip = __builtin_amdgcn_readlane(ip_all, p);
        const int jp = __builtin_amdgcn_readlane(jp_all, p);

        // A = S_p synthesized in-register (rows >= 4 are zero automatically).
        v2f a;
        a.x = ((mrow == k0) ? 1.0f : 0.0f) +
              (((mrow == ip) && (k0 == jp)) ? kv : 0.0f);
        a.y = ((mrow == k1) ? 1.0f : 0.0f) +
              (((mrow == ip) && (k1 == jp)) ? kv : 0.0f);

        v8f c = {};
        // 8 args: (neg_a, A, neg_b, B, c_mod, C, reuse_a, reuse_b)
        d = __builtin_amdgcn_wmma_f32_16x16x4_f32(false, a, false, b,
                                                  (short)0, c, false, false);

        // D -> next B, in-register. D layout: d[r] lane n = M[r][n] (rows 0-7
        // in lanes 0-15). Columns >= 4 of D are already zero (B cols >=4 were
        // zero), so no masking needed.
        const float r0 = bperm_f(ncol, d[0]);   // M[0][ncol]
        const float r1 = bperm_f(ncol, d[1]);   // M[1][ncol]
        const float r2 = bperm_f(ncol, d[2]);   // M[2][ncol]
        const float r3 = bperm_f(ncol, d[3]);   // M[3][ncol]
        b.x = lo ? r0 : r2;                     // row k0
        b.y = lo ? r1 : r3;                     // row k1
    }

    // Final M: rows 0..3 in d[0..3], cols = lane (0..3).
    if (lane < 4) {
        out[ 0 + lane] = d[0];
        out[ 4 + lane] = d[1];
        out[ 8 + lane] = d[2];
        out[12 + lane] = d[3];
    }
}

extern "C" void kernel_launch(void* const* d_in, const int* in_sizes, int n_in,
                              void* d_out, int out_size, void* d_ws, size_t ws_size,
                              hipStream_t stream) {
    const float* k_raw   = (const float*)d_in[0];
    const int*   pairs_i = (const int*)d_in[1];
    const int*   pairs_j = (const int*)d_in[2];
    float*       out     = (float*)d_out;
    (void)in_sizes; (void)n_in; (void)out_size; (void)d_ws; (void)ws_size;

    // Latency-bound problem: exactly one wave32 on one WGP.
    intshear4_wmma_kernel<<<1, 32, 0, stream>>>(k_raw, pairs_i, pairs_j, out);
}